// CTCLoss_45234595562042
// MI455X (gfx1250) — compile-verified
//
#include <hip/hip_runtime.h>
#include <hip/hip_bf16.h>

#define B_N 128
#define C_N 1024
#define T_N 512
#define S_N 64
#define L_N 129          // 2*S+1 extended labels
#define NEGV (-1e9f)

typedef float v2f __attribute__((ext_vector_type(2)));
typedef float v8f __attribute__((ext_vector_type(8)));

#if defined(__has_builtin)
#if __has_builtin(__builtin_amdgcn_wmma_f32_16x16x4_f32)
#define HAVE_WMMA_F32X4 1
#endif
#endif

// -----------------------------------------------------------------------------
// Kernel 1: per-(b,t) log-sum-exp over C classes.
// Wave layout for V_WMMA_F32_16X16X4_F32 A-matrix (16 t-rows x 4 classes):
//   lane = m + 16*khi, m = t-row (M), khi selects K={2,3} half.
//   A VGPR v holds class c0 + 2*khi + v  (ISA 32-bit A layout).
// B = ones(4x16) so D[m,n] += sum_k A[m,k]: WMMA acts as a 64-add reducer that
// runs on the matrix pipe while the VALU does the exp()s.
// Block = 256 threads = 8 waves, each wave owns 16 t's; block owns (b, 128 t's).
// -----------------------------------------------------------------------------
__global__ __launch_bounds__(256) void ctc_lse_kernel(
    const float* __restrict__ preds, float* __restrict__ lse) {
  const int b      = blockIdx.x >> 2;
  const int tchunk = blockIdx.x & 3;
  const int wave   = threadIdx.x >> 5;
  const int lane   = threadIdx.x & 31;
  const int m      = lane & 15;
  const int khi    = lane >> 4;
  const int t      = tchunk * 128 + wave * 16 + m;

  const float* p = preds + (size_t)b * C_N * T_N + t;

  // ---- pass 1: running max over this lane's class subset (c%4 in {2khi,2khi+1})
  float mx = -3.402823466e38f;
#pragma unroll 2
  for (int c0 = 0; c0 < C_N; c0 += 8) {
    const int cA = c0 + 2 * khi;
    float x0 = p[(size_t)(cA    ) * T_N];
    float x1 = p[(size_t)(cA + 1) * T_N];
    float x2 = p[(size_t)(cA + 4) * T_N];
    float x3 = p[(size_t)(cA + 5) * T_N];
    mx = fmaxf(mx, fmaxf(fmaxf(x0, x1), fmaxf(x2, x3)));
  }
  // combine the two K-halves (same t-row lives in lane L and L^16)
  mx = fmaxf(mx, __shfl_xor(mx, 16, 32));

  // ---- pass 2: sum of exp(x - mx), accumulated by WMMA (B = ones)
#if defined(HAVE_WMMA_F32X4)
  v8f acc = {0.f, 0.f, 0.f, 0.f, 0.f, 0.f, 0.f, 0.f};
  v2f bones; bones.x = 1.0f; bones.y = 1.0f;
#pragma unroll 2
  for (int c0 = 0; c0 < C_N; c0 += 4) {
    const int cA = c0 + 2 * khi;
    v2f a;
    a.x = __expf(p[(size_t)(cA    ) * T_N] - mx);
    a.y = __expf(p[(size_t)(cA + 1) * T_N] - mx);
    acc = __builtin_amdgcn_wmma_f32_16x16x4_f32(
        /*neg_a=*/false, a, /*neg_b=*/false, bones,
        /*c_mod=*/(short)0, acc, /*reuse_a=*/false, /*reuse_b=*/false);
  }
  // Extract row sums from the C/D layout: VGPR r -> M=r (lanes 0-15), M=r+8
  // (lanes 16-31); every column holds the same sum since B was all-ones.
  __shared__ float red[8][256];
#pragma unroll
  for (int r = 0; r < 8; r++) red[wave][r * 32 + lane] = acc[r];
  __syncthreads();
  if (lane < 16) {
    float s = (m < 8) ? red[wave][m * 32 + 0] : red[wave][(m - 8) * 32 + 16];
    lse[b * T_N + t] = mx + __logf(s);
  }
#else
  // VALU fallback: per-lane partial sum, then fold K-halves by shuffle.
  float sacc = 0.f;
#pragma unroll 2
  for (int c0 = 0; c0 < C_N; c0 += 4) {
    const int cA = c0 + 2 * khi;
    sacc += __expf(p[(size_t)(cA    ) * T_N] - mx);
    sacc += __expf(p[(size_t)(cA + 1) * T_N] - mx);
  }
  sacc += __shfl_xor(sacc, 16, 32);
  if (lane < 16) lse[b * T_N + t] = mx + __logf(sacc);
#endif
}

// -----------------------------------------------------------------------------
// Kernel 2: CTC alpha recursion. One wave32 per batch element, 5 extended-label
// positions per lane (32*5 >= 129), neighbor alphas across lanes via shfl_up.
// The t-loop has no barriers and runs in registers; readout goes via LDS + one barrier.
// -----------------------------------------------------------------------------
__global__ __launch_bounds__(32) void ctc_dp_kernel(
    const float* __restrict__ preds, const int* __restrict__ gt,
    const int* __restrict__ plen_in, const int* __restrict__ glen_in,
    const float* __restrict__ lse, float* __restrict__ loss_b) {
  const int b    = blockIdx.x;
  const int lane = threadIdx.x;
  const int plen = plen_in[b];
  const int glen = glen_in[b];
  const float* lseb = lse + b * T_N;

  const float* rowp[5];
  float alpha[5];
  bool  act[5], cskip[5];

#pragma unroll
  for (int j = 0; j < 5; j++) {
    const int l = lane * 5 + j;
    act[j] = (l < L_N);
    int  lab = 0;
    bool cs  = false;
    if (act[j] && (l & 1)) {
      lab = gt[b * S_N + (l >> 1)];
      cs  = (l >= 3) && (lab != gt[b * S_N + (l >> 1) - 1]);
    }
    cskip[j] = cs;
    rowp[j]  = preds + (size_t)b * C_N * T_N + (size_t)lab * T_N;
    const float e0 = rowp[j][0] - lseb[0];
    alpha[j] = (act[j] && l <= 1) ? e0 : NEGV;
  }

  for (int t = 1; t < plen; t++) {
    const float lse_t = lseb[t];
    // old boundary alphas from the previous lane (positions l-1, l-2 of j==0/1)
    float pm1 = __shfl_up(alpha[4], 1, 32);
    float pm2 = __shfl_up(alpha[3], 1, 32);
    if (lane == 0) { pm1 = NEGV; pm2 = NEGV; }

    float na[5];
#pragma unroll
    for (int j = 0; j < 5; j++) {
      float a  = alpha[j];
      float a1 = (j == 0) ? pm1 : alpha[j - 1];
      float a2 = (j == 0) ? pm2 : ((j == 1) ? pm1 : alpha[j - 2]);
      if (!cskip[j]) a2 = NEGV;
      const float e  = rowp[j][t] - lse_t;            // row 0 (blank) for idle slots: safe
      const float mm = fmaxf(a, fmaxf(a1, a2));
      const float s  = __expf(a - mm) + __expf(a1 - mm) + __expf(a2 - mm);
      na[j] = act[j] ? (mm + __logf(s) + e) : NEGV;
    }
#pragma unroll
    for (int j = 0; j < 5; j++) alpha[j] = na[j];
  }

  // readout: ll = logaddexp(alpha[2*glen], alpha[2*glen-1])
  __shared__ float afin[160];
#pragma unroll
  for (int j = 0; j < 5; j++) afin[lane * 5 + j] = alpha[j];
  __syncthreads();
  if (lane == 0) {
    const int   idx = 2 * glen;          // in [64,128]
    const float al  = afin[idx];
    const float ap  = afin[idx - 1];
    const float mm  = fmaxf(al, ap);
    const float ll  = mm + __logf(__expf(al - mm) + __expf(ap - mm));
    loss_b[b] = -ll / (float)glen;
  }
}

// -----------------------------------------------------------------------------
// Kernel 3: mean over batch -> scalar output.
// -----------------------------------------------------------------------------
__global__ __launch_bounds__(128) void ctc_reduce_kernel(
    const float* __restrict__ loss_b, float* __restrict__ out) {
  __shared__ float sh[128];
  sh[threadIdx.x] = loss_b[threadIdx.x];
  __syncthreads();
  for (int s = 64; s > 0; s >>= 1) {
    if (threadIdx.x < s) sh[threadIdx.x] += sh[threadIdx.x + s];
    __syncthreads();
  }
  if (threadIdx.x == 0) out[0] = sh[0] / 128.0f;
}

extern "C" void kernel_launch(void* const* d_in, const int* in_sizes, int n_in,
                              void* d_out, int out_size, void* d_ws, size_t ws_size,
                              hipStream_t stream) {
  const float* preds = (const float*)d_in[0];   // (B, C, T) f32
  const int*   gt    = (const int*)d_in[1];     // (B, S)    i32
  const int*   plen  = (const int*)d_in[2];     // (B,)      i32
  const int*   glen  = (const int*)d_in[3];     // (B,)      i32
  float*       out   = (float*)d_out;

  float* lse    = (float*)d_ws;                 // B*T floats
  float* loss_b = lse + (size_t)B_N * T_N;      // B floats

  ctc_lse_kernel<<<B_N * 4, 256, 0, stream>>>(preds, lse);
  ctc_dp_kernel<<<B_N, 32, 0, stream>>>(preds, gt, plen, glen, lse, loss_b);
  ctc_reduce_kernel<<<1, 128, 0, stream>>>(loss_b, out);
}